// Gmm_63067299774561
// MI455X (gfx1250) — compile-verified
//
#include <hip/hip_runtime.h>
#include <hip/hip_bf16.h>

// GMM EM step for MI455X (gfx1250, wave32).
// d_in: x[N*8] f32, mu[K*8] f32, var[K*64] f32, Pi[K] f32
// d_out: W[N*K] f32 followed by loss[1] f32
// d_ws: Bmat[K*80] | Bmat2[K*80] | S[K*80]  (floats)

#define GN 262144
#define GK 512
#define GD 8
#define BSTR 80          // packed B row stride (76 used, padded to 80)
#define LOG2PI_F 1.8378770664093453f
#define EPS_F 1e-6f

typedef float v2f __attribute__((ext_vector_type(2)));
typedef float v8f __attribute__((ext_vector_type(8)));

// ---------------------------------------------------------------------------
// Per-cluster packing: Cholesky -> logdet -> inverse -> B row
//   Brow[j<64]  = -0.5 * P[i][jj]   (j = i*8+jj)
//   Brow[64+i]  = (P mu)[i]
//   Brow[72]    = logPi - 0.5*(D*log2pi + logdet + mu^T P mu)
//   Brow[73..79]= 0
// ---------------------------------------------------------------------------
__device__ __forceinline__ void pack_cluster(const float* V, const float* m8,
                                             float logPi, float* Brow) {
    float L[64];
    float logdet = 0.0f;
#pragma unroll
    for (int j = 0; j < 8; ++j) {
        float s = V[j * 8 + j];
#pragma unroll
        for (int p = 0; p < 8; ++p) if (p < j) s -= L[j * 8 + p] * L[j * 8 + p];
        float d = sqrtf(fmaxf(s, 1e-20f));
        L[j * 8 + j] = d;
        logdet += 2.0f * logf(d);
        float dinv = 1.0f / d;
#pragma unroll
        for (int i = 0; i < 8; ++i) if (i > j) {
            float t = V[i * 8 + j];
#pragma unroll
            for (int p = 0; p < 8; ++p) if (p < j) t -= L[i * 8 + p] * L[j * 8 + p];
            L[i * 8 + j] = t * dinv;
        }
    }
    // Li = L^{-1} (lower triangular)
    float Li[64];
#pragma unroll
    for (int j = 0; j < 8; ++j) {
        Li[j * 8 + j] = 1.0f / L[j * 8 + j];
#pragma unroll
        for (int i = 0; i < 8; ++i) if (i > j) {
            float t = 0.0f;
#pragma unroll
            for (int p = 0; p < 8; ++p) if (p >= j && p < i) t += L[i * 8 + p] * Li[p * 8 + j];
            Li[i * 8 + j] = -t / L[i * 8 + i];
        }
    }
    // P = Li^T Li ; Pmu ; mu^T P mu
    float cst = 0.0f;
#pragma unroll
    for (int i = 0; i < 8; ++i) {
        float pm = 0.0f;
#pragma unroll
        for (int j = 0; j < 8; ++j) {
            float t = 0.0f;
            const int p0 = (i > j) ? i : j;
#pragma unroll
            for (int p = 0; p < 8; ++p) if (p >= p0) t += Li[p * 8 + i] * Li[p * 8 + j];
            Brow[i * 8 + j] = -0.5f * t;
            pm += t * m8[j];
        }
        Brow[64 + i] = pm;
        cst += pm * m8[i];
    }
    Brow[72] = logPi - 0.5f * (8.0f * LOG2PI_F + logdet + cst);
#pragma unroll
    for (int j = 73; j < 80; ++j) Brow[j] = 0.0f;
}

__global__ void zero_kernel(float* __restrict__ S, float* __restrict__ loss, int n) {
    int i = blockIdx.x * blockDim.x + threadIdx.x;
    if (i < n) S[i] = 0.0f;
    if (i == 0) *loss = 0.0f;
}

__global__ void prep_from_params(const float* __restrict__ mu, const float* __restrict__ var,
                                 const float* __restrict__ Pi, float* __restrict__ Bmat) {
    int k = blockIdx.x * blockDim.x + threadIdx.x;
    if (k >= GK) return;
    float V[64], m8[8];
#pragma unroll
    for (int i = 0; i < 64; ++i) V[i] = var[(size_t)k * 64 + i];
#pragma unroll
    for (int i = 0; i < 8; ++i) m8[i] = mu[(size_t)k * 8 + i];
    pack_cluster(V, m8, logf(Pi[k]), Bmat + (size_t)k * BSTR);
}

// M-step finalize: stats S[k][0..63]=Sxx, [64..71]=Sx, [72]=Nk -> new params -> B'
__global__ void prep_from_stats(const float* __restrict__ S, float* __restrict__ Bmat2) {
    int k = blockIdx.x * blockDim.x + threadIdx.x;
    if (k >= GK) return;
    const float* Sk = S + (size_t)k * BSTR;
    float Nk = fmaxf(Sk[72], 1e-12f);
    float inv = 1.0f / Nk;
    float m8[8];
#pragma unroll
    for (int i = 0; i < 8; ++i) m8[i] = Sk[64 + i] * inv;
    float V[64];
#pragma unroll
    for (int i = 0; i < 8; ++i)
#pragma unroll
        for (int j = 0; j < 8; ++j)
            V[i * 8 + j] = Sk[i * 8 + j] * inv - m8[i] * m8[j] + ((i == j) ? EPS_F : 0.0f);
    float logPi = logf(Nk * (1.0f / (float)GN));
    pack_cluster(V, m8, logPi, Bmat2 + (size_t)k * BSTR);
}

// ---------------------------------------------------------------------------
// E-step / loss kernel: logp[16 rows][512 k] via V_WMMA_F32_16X16X4_F32.
// Block = 256 threads (8 waves); wave w owns k-tiles 4w..4w+3.
// A lane layout (16x4 f32): m = lane&15 (x row), k-elem = 2*(lane>>4)+vgpr.
// B lane layout (4x16):     col = lane&15 (cluster), row = 2*(lane>>4)+vgpr.
// D lane layout (16x16):    m = vgpr + 8*(lane>>4), col = lane&15.
// mode 0: softmax -> write W.  mode 1: logsumexp -> atomicAdd(mean) to loss.
// ---------------------------------------------------------------------------
__global__ __launch_bounds__(256) void estep_kernel(const float* __restrict__ x,
                                                    const float* __restrict__ B,
                                                    float* __restrict__ Wout,
                                                    float* __restrict__ loss,
                                                    int mode) {
    __shared__ float s_logp[16 * 512];
    __shared__ float s_pmax[256];
    __shared__ float s_rmax[16];
    __shared__ float s_psum[256];
    __shared__ float s_rsum[16];

    const int tid = threadIdx.x;
    const int lane = tid & 31;
    const int c = lane & 15;
    const int hi = lane >> 4;       // 0/1: which half of the K window this lane holds
    const int w = tid >> 5;
    const int n0 = blockIdx.x * 16;

    // this lane's x row (rows duplicated across half-waves)
    float xr[8];
    const float* xp = x + (size_t)(n0 + c) * GD;
#pragma unroll
    for (int i = 0; i < 8; ++i) xr[i] = xp[i];

    v8f acc[4];
#pragma unroll
    for (int t = 0; t < 4; ++t) acc[t] = (v8f){};

#pragma unroll
    for (int kk = 0; kk < 76; kk += 4) {
        // features j = kk + 2*hi + {0,1}; for j<64: f = x[j>>3]*x[j&7] and
        // (j>>3) is constant over the 4-wide window, so indices fold to consts.
        v2f a;
        if (kk < 64) {
            const float lead = xr[kk >> 3];
            const float s0 = hi ? xr[(kk & 7) + 2] : xr[(kk & 7) + 0];
            const float s1 = hi ? xr[(kk & 7) + 3] : xr[(kk & 7) + 1];
            a.x = lead * s0;
            a.y = lead * s1;
        } else if (kk < 72) {
            a.x = hi ? xr[kk - 62] : xr[kk - 64];
            a.y = hi ? xr[kk - 61] : xr[kk - 63];
        } else {                      // j==72 -> 1, j>72 -> 0
            a.x = hi ? 0.0f : 1.0f;
            a.y = 0.0f;
        }
#pragma unroll
        for (int t = 0; t < 4; ++t) {
            const int k0 = (w * 4 + t) * 16;
            v2f b = *(const v2f*)(B + (size_t)(k0 + c) * BSTR + kk + 2 * hi);
            acc[t] = __builtin_amdgcn_wmma_f32_16x16x4_f32(
                false, a, false, b, (short)0, acc[t], false, false);
        }
    }

    // spill logp tiles to LDS: row = v + 8*hi, col = k-tile*16 + c
#pragma unroll
    for (int t = 0; t < 4; ++t)
#pragma unroll
        for (int v = 0; v < 8; ++v)
            s_logp[(v + 8 * hi) * 512 + (w * 4 + t) * 16 + c] = acc[t][v];
    __syncthreads();

    // row-wise softmax / logsumexp over 512 columns
    const int r = tid >> 4, seg = tid & 15;
    float mx = -3.0e38f;
#pragma unroll 4
    for (int i = 0; i < 32; ++i) mx = fmaxf(mx, s_logp[r * 512 + seg * 32 + i]);
    s_pmax[tid] = mx;
    __syncthreads();
    if (tid < 16) {
        float m2 = -3.0e38f;
        for (int i = 0; i < 16; ++i) m2 = fmaxf(m2, s_pmax[tid * 16 + i]);
        s_rmax[tid] = m2;
    }
    __syncthreads();
    const float rm = s_rmax[r];
    float sum = 0.0f;
#pragma unroll 4
    for (int i = 0; i < 32; ++i) {
        const int idx = r * 512 + seg * 32 + i;
        const float e = __expf(s_logp[idx] - rm);
        sum += e;
        s_logp[idx] = e;
    }
    s_psum[tid] = sum;
    __syncthreads();
    if (tid < 16) {
        float s2 = 0.0f;
        for (int i = 0; i < 16; ++i) s2 += s_psum[tid * 16 + i];
        s_rsum[tid] = s2;
    }
    __syncthreads();

    if (mode == 0) {
#pragma unroll
        for (int i = 0; i < 32; ++i) {
            const int idx = tid + 256 * i;
            const int row = idx >> 9, col = idx & 511;
            Wout[(size_t)(n0 + row) * GK + col] = s_logp[idx] / s_rsum[row];
        }
    } else {
        if (tid == 0) {
            float accl = 0.0f;
            for (int r2 = 0; r2 < 16; ++r2) accl += s_rmax[r2] + __logf(s_rsum[r2]);
            atomicAdd(loss, accl * (1.0f / (float)GN));
        }
    }
}

// ---------------------------------------------------------------------------
// M-step GEMM: S[k][j] += sum_n W[n][k] * f(n,j), j<80 (f73..79 == 0).
// grid (32 k-tiles, 64 n-chunks of 4096). D-tile = [16 clusters x 16 features].
// A = W^T: a lane m = lane&15 -> cluster k0+c; k-elem = rows n0+2*hi+{0,1}.
// B rows = x-feature rows for the same 4 n's; feature col j = t*16 + c.
// ---------------------------------------------------------------------------
__global__ __launch_bounds__(256) void mstep_kernel(const float* __restrict__ x,
                                                    const float* __restrict__ Wm,
                                                    float* __restrict__ S) {
    __shared__ float red[8 * 1280];
    const int tid = threadIdx.x;
    const int lane = tid & 31;
    const int c = lane & 15;
    const int hi = lane >> 4;
    const int w = tid >> 5;
    const int k0 = blockIdx.x * 16;
    const int chunkBase = blockIdx.y * 4096;
    const int c7 = c & 7;
    const int ch = c >> 3;   // 0 or 1

    v8f acc[5];
#pragma unroll
    for (int t = 0; t < 5; ++t) acc[t] = (v8f){};

    for (int it = 0; it < 128; ++it) {
        const int n0 = chunkBase + (it * 8 + w) * 4;
        const int ra = n0 + 2 * hi;
        const int rb = ra + 1;
        const float* xa = x + (size_t)ra * GD;
        const float* xb = x + (size_t)rb * GD;
        const float ua = xa[c7], ub = xb[c7];
        // feature tile 4: j = 64+c -> x[c] (c<8), 1 (c==8), 0 (c>8); x[c]==ua for c<8
        const float f4a = (c < 8) ? ua : ((c == 8) ? 1.0f : 0.0f);
        const float f4b = (c < 8) ? ub : ((c == 8) ? 1.0f : 0.0f);

        v2f a;
        a.x = Wm[(size_t)ra * GK + k0 + c];
        a.y = Wm[(size_t)rb * GK + k0 + c];

#pragma unroll
        for (int t = 0; t < 4; ++t) {    // j = t*16+c < 64 : f = x[2t+(c>>3)] * x[c&7]
            v2f b;
            b.x = xa[2 * t + ch] * ua;
            b.y = xb[2 * t + ch] * ub;
            acc[t] = __builtin_amdgcn_wmma_f32_16x16x4_f32(
                false, a, false, b, (short)0, acc[t], false, false);
        }
        {
            v2f b; b.x = f4a; b.y = f4b;
            acc[4] = __builtin_amdgcn_wmma_f32_16x16x4_f32(
                false, a, false, b, (short)0, acc[4], false, false);
        }
    }

    // LDS reduce 8 waves, then ~1.3K fp32 atomics into global S
#pragma unroll
    for (int t = 0; t < 5; ++t)
#pragma unroll
        for (int v = 0; v < 8; ++v)
            red[w * 1280 + t * 256 + (v + 8 * hi) * 16 + c] = acc[t][v];
    __syncthreads();
#pragma unroll
    for (int i = 0; i < 5; ++i) {
        const int e = tid + 256 * i;
        float s = 0.0f;
#pragma unroll
        for (int ww = 0; ww < 8; ++ww) s += red[ww * 1280 + e];
        const int t = e >> 8, m = (e >> 4) & 15, cc = e & 15;
        atomicAdd(&S[(size_t)(k0 + m) * BSTR + t * 16 + cc], s);
    }
}

extern "C" void kernel_launch(void* const* d_in, const int* in_sizes, int n_in,
                              void* d_out, int out_size, void* d_ws, size_t ws_size,
                              hipStream_t stream) {
    (void)in_sizes; (void)n_in; (void)out_size; (void)ws_size;
    const float* x   = (const float*)d_in[0];
    const float* mu  = (const float*)d_in[1];
    const float* var = (const float*)d_in[2];
    const float* Pi  = (const float*)d_in[3];

    float* Wout = (float*)d_out;
    float* loss = Wout + (size_t)GN * GK;

    float* Bmat  = (float*)d_ws;                 // [512*80]
    float* Bmat2 = Bmat + GK * BSTR;             // [512*80]
    float* S     = Bmat2 + GK * BSTR;            // [512*80]

    zero_kernel<<<(GK * BSTR + 255) / 256, 256, 0, stream>>>(S, loss, GK * BSTR);
    prep_from_params<<<2, 256, 0, stream>>>(mu, var, Pi, Bmat);
    estep_kernel<<<GN / 16, 256, 0, stream>>>(x, Bmat, Wout, loss, 0);
    mstep_kernel<<<dim3(32, 64), 256, 0, stream>>>(x, Wout, S);
    prep_from_stats<<<2, 256, 0, stream>>>(S, Bmat2);
    estep_kernel<<<GN / 16, 256, 0, stream>>>(x, Bmat2, Wout, loss, 1);
}